// GATConv_wrapper_75900662055241
// MI455X (gfx1250) — compile-verified
//
#include <hip/hip_runtime.h>
#include <hip/hip_bf16.h>

typedef float v2f __attribute__((ext_vector_type(2)));
typedef float v8f __attribute__((ext_vector_type(8)));

#define NEG_SLOPE 0.2f

// ---------------------------------------------------------------------------
// Fragment layouts for V_WMMA_F32_16X16X4_F32 (wave32):
//   A (16x4): lane l -> row M = l&15;  VGPR0 = K = k+2*(l>>4), VGPR1 = K+1
//   B (4x16): lane l -> col N = l&15;  VGPR0 = K row k+2*(l>>4), VGPR1 = K+1
//   D (16x16): VGPR r: lanes 0-15 -> M=r, lanes 16-31 -> M=r+8, N = l&15
// We pre-pack A and B fragments so the GEMM inner loop is pure b64-load + wmma.
// ---------------------------------------------------------------------------

// Pack x[N,K] into Apack[mTile][kstep][lane] (float2 per lane)
__global__ void gat_pack_A(const float* __restrict__ x, float* __restrict__ Apack,
                           int N, int K, int nMTiles, int nKSteps)
{
    const int t = blockIdx.x * blockDim.x + threadIdx.x;
    const int total = nMTiles * nKSteps * 32;
    if (t >= total) return;
    const int lane = t & 31;
    const int ks   = (t >> 5) % nKSteps;
    const int tile = (t >> 5) / nKSteps;
    const int lo = lane & 15, hi = lane >> 4;
    int row = tile * 16 + lo;
    if (row >= N) row = N - 1;            // clamp; duplicate rows are never stored
    const int ka = ks * 4 + 2 * hi;
    float2 v;
    v.x = x[(size_t)row * K + ka];
    v.y = x[(size_t)row * K + ka + 1];
    ((float2*)Apack)[t] = v;
}

// Pack W[K,F] into Bpack[nTile][kstep][lane] (float2 per lane), zero-padded cols
__global__ void gat_pack_B(const float* __restrict__ W, float* __restrict__ Bpack,
                           int K, int F, int nNTiles, int nKSteps)
{
    const int t = blockIdx.x * blockDim.x + threadIdx.x;
    const int total = nNTiles * nKSteps * 32;
    if (t >= total) return;
    const int lane = t & 31;
    const int ks   = (t >> 5) % nKSteps;
    const int tile = (t >> 5) / nKSteps;
    const int lo = lane & 15, hi = lane >> 4;
    const int col = tile * 16 + lo;
    const int ka = ks * 4 + 2 * hi;
    float2 v;
    v.x = (col < F) ? W[(size_t)(ka    ) * F + col] : 0.f;
    v.y = (col < F) ? W[(size_t)(ka + 1) * F + col] : 0.f;
    ((float2*)Bpack)[t] = v;
}

// ---------------------------------------------------------------------------
// K1: h = x @ W. One wave32 per 16x16 tile; branch-free unrolled k-loop.
// ---------------------------------------------------------------------------
__global__ __launch_bounds__(256) void gat_gemm_wmma(
    const float* __restrict__ Apack,
    const float* __restrict__ Bpack,
    float* __restrict__ h,
    int N, int F, int nMTiles, int nNTiles, int nKSteps)
{
    const int wave = (blockIdx.x * blockDim.x + threadIdx.x) >> 5;
    const int lane = threadIdx.x & 31;
    if (wave >= nMTiles * nNTiles) return;
    const int nTile = wave % nNTiles;
    const int mTile = wave / nNTiles;

    const v2f* __restrict__ Ap = (const v2f*)Apack + (size_t)mTile * nKSteps * 32 + lane;
    const v2f* __restrict__ Bp = (const v2f*)Bpack + (size_t)nTile * nKSteps * 32 + lane;

    v8f c = {0.f, 0.f, 0.f, 0.f, 0.f, 0.f, 0.f, 0.f};

#pragma unroll
    for (int ks = 0; ks < 25; ++ks) {      // K=100 / 4
        const v2f a = Ap[ks * 32];
        const v2f b = Bp[ks * 32];
        c = __builtin_amdgcn_wmma_f32_16x16x4_f32(
                false, a, false, b, (short)0, c, false, false);
    }

    const int lo = lane & 15, hi = lane >> 4;
    const int col = nTile * 16 + lo;
    if (col < F) {
#pragma unroll
        for (int r = 0; r < 8; ++r) {
            const int m = mTile * 16 + r + 8 * hi;
            if (m < N) h[(size_t)m * F + col] = c[r];
        }
    }
}

// ---------------------------------------------------------------------------
// K2: per node (one wave each): a_s = h.att_src, a_d = h.att_dst,
//     init m = -inf, denom = 0, zero out_accum row.
// ---------------------------------------------------------------------------
__global__ void gat_node_prep(
    const float* __restrict__ h,
    const float* __restrict__ att_src,
    const float* __restrict__ att_dst,
    float* __restrict__ a_s, float* __restrict__ a_d,
    float* __restrict__ mmax, float* __restrict__ denom,
    float* __restrict__ out_accum,
    int N, int F)
{
    const int wave = (blockIdx.x * blockDim.x + threadIdx.x) >> 5;
    const int lane = threadIdx.x & 31;
    if (wave >= N) return;

    const float* hr = h + (size_t)wave * F;
    float* orow = out_accum + (size_t)wave * F;

    float s = 0.f, d = 0.f;
    for (int f = lane; f < F; f += 32) {
        const float v = hr[f];
        s += v * att_src[f];
        d += v * att_dst[f];
        orow[f] = 0.f;
    }
    for (int o = 16; o > 0; o >>= 1) {
        s += __shfl_down(s, o, 32);
        d += __shfl_down(d, o, 32);
    }
    if (lane == 0) {
        a_s[wave]   = s;
        a_d[wave]   = d;
        mmax[wave]  = -__builtin_inff();
        denom[wave] = 0.f;
    }
}

// float atomic max via signed-max / unsigned-min bit trick (init = -inf)
__device__ __forceinline__ void atomicMaxFloat(float* addr, float val) {
    if (val >= 0.f)
        atomicMax((int*)addr, __float_as_int(val));
    else
        atomicMin((unsigned int*)addr, __float_as_uint(val));
}

// edge i in [0, E): src = ei[i], dst = ei[E+i]; i in [E, E+N): self loop
__device__ __forceinline__ void edge_src_dst(const int* __restrict__ ei,
                                             int i, int E, int& src, int& dst) {
    if (i < E) { src = ei[i]; dst = ei[E + i]; }
    else       { src = i - E; dst = i - E; }
}

// ---------------------------------------------------------------------------
// K3: segment max of leaky-relu logits over destination nodes.
// ---------------------------------------------------------------------------
__global__ void gat_edge_max(
    const int* __restrict__ ei,
    const float* __restrict__ a_s, const float* __restrict__ a_d,
    float* __restrict__ mmax, int E, int N)
{
    const int i = blockIdx.x * blockDim.x + threadIdx.x;
    if (i >= E + N) return;
    int src, dst;
    edge_src_dst(ei, i, E, src, dst);
    float e = a_s[src] + a_d[dst];
    e = (e > 0.f) ? e : NEG_SLOPE * e;
    atomicMaxFloat(&mmax[dst], e);
}

// ---------------------------------------------------------------------------
// K4: ex = exp(e - m[dst]); denom[dst] += ex; stash ex per edge.
// ---------------------------------------------------------------------------
__global__ void gat_edge_expsum(
    const int* __restrict__ ei,
    const float* __restrict__ a_s, const float* __restrict__ a_d,
    const float* __restrict__ mmax,
    float* __restrict__ denom, float* __restrict__ exbuf,
    int E, int N)
{
    const int i = blockIdx.x * blockDim.x + threadIdx.x;
    if (i >= E + N) return;
    int src, dst;
    edge_src_dst(ei, i, E, src, dst);
    float e = a_s[src] + a_d[dst];
    e = (e > 0.f) ? e : NEG_SLOPE * e;
    const float ex = __expf(e - mmax[dst]);
    exbuf[i] = ex;
    atomicAdd(&denom[dst], ex);
}

// ---------------------------------------------------------------------------
// K5: out_accum[dst,:] += (ex/denom[dst]) * h[src,:]  (one wave per edge).
//     h + out_accum (80MB) fit in the 192MB L2 -> scatter runs at L2 rate.
//     float4 loads (global_load_b128) for the gathered h row.
// ---------------------------------------------------------------------------
__global__ void gat_scatter(
    const int* __restrict__ ei,
    const float* __restrict__ h,
    const float* __restrict__ exbuf,
    const float* __restrict__ denom,
    float* __restrict__ out_accum,
    int E, int N, int F)
{
    const int wave = (blockIdx.x * blockDim.x + threadIdx.x) >> 5;
    const int lane = threadIdx.x & 31;
    if (wave >= E + N) return;
    int src, dst;
    edge_src_dst(ei, wave, E, src, dst);
    const float alpha = exbuf[wave] / denom[dst];
    const float4* hs4 = (const float4*)(h + (size_t)src * F);
    float* od = out_accum + (size_t)dst * F;
    const int nC = F >> 2;                     // 50 float4 chunks
    for (int i = lane; i < nC; i += 32) {
        const float4 v = hs4[i];
        atomicAdd(&od[4 * i + 0], alpha * v.x);
        atomicAdd(&od[4 * i + 1], alpha * v.y);
        atomicAdd(&od[4 * i + 2], alpha * v.z);
        atomicAdd(&od[4 * i + 3], alpha * v.w);
    }
    for (int f = (nC << 2) + lane; f < F; f += 32)   // tail (F%4, unused here)
        atomicAdd(&od[f], alpha * h[(size_t)src * F + f]);
}

// ---------------------------------------------------------------------------
// K6a: zero the column-partial buffer.
// ---------------------------------------------------------------------------
__global__ void gat_zero_cols(float* __restrict__ col_part, int F)
{
    const int t = blockIdx.x * blockDim.x + threadIdx.x;
    if (t < F) col_part[t] = 0.f;
}

// K6b: each block owns a row-chunk; thread t owns column t (coalesced reads).
__global__ void gat_colsum(
    const float* __restrict__ out_accum,
    float* __restrict__ col_part,
    int N, int F, int rowsPerBlock)
{
    const int col = threadIdx.x;
    if (col >= F) return;
    const int r0 = blockIdx.x * rowsPerBlock;
    const int r1 = min(r0 + rowsPerBlock, N);
    float s = 0.f;
    for (int r = r0; r < r1; ++r)
        s += out_accum[(size_t)r * F + col];
    atomicAdd(&col_part[col], s);
}

// K6c: finalize d_out = bias + col_part / N  (no accumulation across replays)
__global__ void gat_finalize(
    const float* __restrict__ col_part,
    const float* __restrict__ bias,
    float* __restrict__ out, int N, int F)
{
    const int t = blockIdx.x * blockDim.x + threadIdx.x;
    if (t < F) out[t] = bias[t] + col_part[t] / (float)N;
}

// ---------------------------------------------------------------------------
extern "C" void kernel_launch(void* const* d_in, const int* in_sizes, int n_in,
                              void* d_out, int out_size, void* d_ws, size_t ws_size,
                              hipStream_t stream) {
    const float* x       = (const float*)d_in[0];
    const int*   ei      = (const int*)  d_in[1];
    const float* W       = (const float*)d_in[2];
    const float* att_src = (const float*)d_in[3];
    const float* att_dst = (const float*)d_in[4];
    const float* bias    = (const float*)d_in[5];
    float* out = (float*)d_out;

    const int F = in_sizes[3];            // 200
    const int K = in_sizes[2] / F;        // 100
    const int N = in_sizes[0] / K;        // 50000
    const int E = in_sizes[1] / 2;        // 800000
    const int M = E + N;                  // edges incl. self loops

    const int nMTiles = (N + 15) / 16;    // 3125
    const int nNTiles = (F + 15) / 16;    // 13
    const int nKSteps = K / 4;            // 25

    // workspace layout (floats)
    float* ws        = (float*)d_ws;
    float* h         = ws;                                   // N*F
    float* a_s       = h     + (size_t)N * F;                // N
    float* a_d       = a_s   + N;                            // N
    float* mmax      = a_d   + N;                            // N
    float* denom     = mmax  + N;                            // N
    float* exbuf     = denom + N;                            // M
    float* out_accum = exbuf + M;                            // N*F
    float* Apack     = out_accum + (size_t)N * F;            // nMTiles*nKSteps*64
    float* Bpack     = Apack + (size_t)nMTiles * nKSteps * 64; // nNTiles*nKSteps*64
    float* col_part  = Bpack + (size_t)nNTiles * nKSteps * 64; // F

    // K0: pack A and B fragments into WMMA per-lane layout
    {
        const int totA = nMTiles * nKSteps * 32;
        gat_pack_A<<<(totA + 255) / 256, 256, 0, stream>>>(x, Apack, N, K, nMTiles, nKSteps);
        const int totB = nNTiles * nKSteps * 32;
        gat_pack_B<<<(totB + 255) / 256, 256, 0, stream>>>(W, Bpack, K, F, nNTiles, nKSteps);
    }

    // K1: WMMA GEMM, one wave per 16x16 tile, 8 waves/block
    {
        const long long waves = (long long)nMTiles * nNTiles;
        const int blocks = (int)((waves * 32 + 255) / 256);
        gat_gemm_wmma<<<blocks, 256, 0, stream>>>(Apack, Bpack, h, N, F,
                                                  nMTiles, nNTiles, nKSteps);
    }

    // K2: per-node logits + init (wave per node)
    {
        const int blocks = (N * 32 + 255) / 256;
        gat_node_prep<<<blocks, 256, 0, stream>>>(
            h, att_src, att_dst, a_s, a_d, mmax, denom, out_accum, N, F);
    }

    // K3: segment max (thread per edge)
    gat_edge_max<<<(M + 255) / 256, 256, 0, stream>>>(ei, a_s, a_d, mmax, E, N);

    // K4: exp + segment sum
    gat_edge_expsum<<<(M + 255) / 256, 256, 0, stream>>>(
        ei, a_s, a_d, mmax, denom, exbuf, E, N);

    // K5: weighted scatter-add (wave per edge)
    {
        const long long totalThreads = (long long)M * 32;
        const int blocks = (int)((totalThreads + 255) / 256);
        gat_scatter<<<blocks, 256, 0, stream>>>(
            ei, h, exbuf, denom, out_accum, E, N, F);
    }

    // K6: column mean + bias
    {
        gat_zero_cols<<<(F + 255) / 256, 256, 0, stream>>>(col_part, F);
        const int rowsPerBlock = 256;
        const int blocks = (N + rowsPerBlock - 1) / rowsPerBlock;
        gat_colsum<<<blocks, 256, 0, stream>>>(out_accum, col_part, N, F, rowsPerBlock);
        gat_finalize<<<(F + 255) / 256, 256, 0, stream>>>(col_part, bias, out, N, F);
    }
}